// ClusterLoss_472446403063
// MI455X (gfx1250) — compile-verified
//
#include <hip/hip_runtime.h>

#define MBLK   64
#define KDIM   256
#define TROWS  4096
#define SPLIT  4
#define NW     8                         // waves per workgroup (256 threads, wave32)
#define RPW    (TROWS / (SPLIT * NW))    // 128 rows per wave
#define ROWLEN (MBLK * KDIM)             // 16384 floats per T-row

typedef __attribute__((ext_vector_type(2))) float v2f;
typedef __attribute__((ext_vector_type(8))) float v8f;

// ---------------------------------------------------------------------------
// Kernel 1: one streaming pass over the 256 MiB input.
// Wave handles fixed block m, a contiguous range of t.  Per (t,m) row:
// coalesced 2x float4 loads (512B per wave per load), butterfly max/S/W,
// entropy accumulated per wave; raw values accumulated into per-lane column
// sums for the L2 batch-mean.  Deterministic LDS tree combine (no atomics).
// ---------------------------------------------------------------------------
__global__ __launch_bounds__(256) void entropy_stream_kernel(
    const float* __restrict__ x,
    float* __restrict__ colsum,   // [SPLIT][MBLK][KDIM]
    float* __restrict__ entp)     // [SPLIT][MBLK]
{
  __shared__ float lcol[NW][KDIM];
  __shared__ float lent[NW];

  const int m     = blockIdx.x / SPLIT;
  const int split = blockIdx.x % SPLIT;
  const int w     = threadIdx.x >> 5;
  const int lane  = threadIdx.x & 31;

  const size_t t0 = (size_t)(split * NW + w) * RPW;
  const float* p  = x + t0 * ROWLEN + (size_t)m * KDIM + lane * 4;

  float c0=0,c1=0,c2=0,c3=0,c4=0,c5=0,c6=0,c7=0;
  float ent = 0.0f;

  for (int r = 0; r < RPW; ++r) {
    const float4 a = *(const float4*)(p);
    const float4 b = *(const float4*)(p + 128);
    if (r + 2 < RPW) {                       // software read-ahead, 64KB stride
      __builtin_prefetch(p + 2 * ROWLEN, 0, 1);
      __builtin_prefetch(p + 2 * ROWLEN + 128, 0, 1);
    }

    // --- row max (wave32 butterfly) ---
    float mx = fmaxf(fmaxf(fmaxf(a.x, a.y), fmaxf(a.z, a.w)),
                     fmaxf(fmaxf(b.x, b.y), fmaxf(b.z, b.w)));
    #pragma unroll
    for (int i = 16; i >= 1; i >>= 1)
      mx = fmaxf(mx, __shfl_xor(mx, i, 32));

    // --- S = sum exp(d), W = sum d*exp(d) ---
    const float d0=a.x-mx, d1=a.y-mx, d2=a.z-mx, d3=a.w-mx;
    const float d4=b.x-mx, d5=b.y-mx, d6=b.z-mx, d7=b.w-mx;
    const float e0=__expf(d0), e1=__expf(d1), e2=__expf(d2), e3=__expf(d3);
    const float e4=__expf(d4), e5=__expf(d5), e6=__expf(d6), e7=__expf(d7);
    float s  = ((e0+e1)+(e2+e3)) + ((e4+e5)+(e6+e7));
    float wv = ((d0*e0+d1*e1)+(d2*e2+d3*e3)) + ((d4*e4+d5*e5)+(d6*e6+d7*e7));
    #pragma unroll
    for (int i = 16; i >= 1; i >>= 1) {
      s  += __shfl_xor(s,  i, 32);
      wv += __shfl_xor(wv, i, 32);
    }
    ent += __logf(s) - wv / s;               // entropy = logS - W/S (uniform in wave)

    // --- column-sum accumulation for batch mean ---
    c0+=a.x; c1+=a.y; c2+=a.z; c3+=a.w;
    c4+=b.x; c5+=b.y; c6+=b.z; c7+=b.w;
    p += ROWLEN;
  }

  lcol[w][lane*4+0]=c0; lcol[w][lane*4+1]=c1; lcol[w][lane*4+2]=c2; lcol[w][lane*4+3]=c3;
  lcol[w][128+lane*4+0]=c4; lcol[w][128+lane*4+1]=c5; lcol[w][128+lane*4+2]=c6; lcol[w][128+lane*4+3]=c7;
  if (lane == 0) lent[w] = ent;
  __syncthreads();

  {
    const int k = threadIdx.x;               // 256 threads <-> 256 columns
    float sum = 0.0f;
    #pragma unroll
    for (int ww = 0; ww < NW; ++ww) sum += lcol[ww][k];
    colsum[((size_t)split * MBLK + m) * KDIM + k] = sum;
  }
  if (threadIdx.x == 0) {
    float sum = 0.0f;
    #pragma unroll
    for (int ww = 0; ww < NW; ++ww) sum += lent[ww];
    entp[split * MBLK + m] = sum;
  }
}

// ---------------------------------------------------------------------------
// Kernel 2: combine split column sums -> block_mean [64][256] in LDS, then
// compute per-block entropy.  The K-dim sums use V_WMMA_F32_16X16X4_F32 with
// a ones B-matrix: D = A(16 rows x 4 k) * ones + C accumulates 16 row sums
// per instruction.  Lane layout matches the ISA 16x4 f32 A-matrix:
// lanes 0-15 = rows, VGPR0/1 = K0/K1; lanes 16-31 = K2/K3.
// ---------------------------------------------------------------------------
__global__ __launch_bounds__(128) void blockmean_wmma_kernel(
    const float* __restrict__ colsum,
    const float* __restrict__ entp,
    float* __restrict__ out)
{
  __shared__ float bm[MBLK][KDIM + 1];       // +1 pad: avoid 16-way bank conflict
  __shared__ float hpart[4];

  for (int i = threadIdx.x; i < MBLK * KDIM; i += 128) {
    float s = 0.0f;
    #pragma unroll
    for (int sp = 0; sp < SPLIT; ++sp) s += colsum[(size_t)sp * MBLK * KDIM + i];
    bm[i >> 8][i & 255] = s * (1.0f / TROWS);
  }
  __syncthreads();

  const int w    = threadIdx.x >> 5;         // wave 0..3 -> rows w*16 .. w*16+15
  const int lane = threadIdx.x & 31;
  const int r    = lane & 15;                // A-matrix row within wave
  const int hk   = lane >> 4;                // 0: K0/K1, 1: K2/K3 of each chunk
  const int m    = w * 16 + r;

  // pass 1: row max
  float mx = -3.402823466e38f;
  for (int c = 0; c < 64; ++c) {
    const int k = c * 4 + hk * 2;
    mx = fmaxf(mx, fmaxf(bm[m][k], bm[m][k + 1]));
  }
  mx = fmaxf(mx, __shfl_xor(mx, 16, 32));    // combine the two half-K lanes

  // pass 2: WMMA-accumulated S and W (B = ones -> row sums in every column)
  v8f accS = {0,0,0,0,0,0,0,0};
  v8f accW = {0,0,0,0,0,0,0,0};
  const v2f ones = {1.0f, 1.0f};
  for (int c = 0; c < 64; ++c) {
    const int k  = c * 4 + hk * 2;
    const float d0 = bm[m][k]     - mx;
    const float d1 = bm[m][k + 1] - mx;
    const float e0 = __expf(d0), e1 = __expf(d1);
    const v2f Ae = {e0, e1};
    const v2f Aw = {d0 * e0, d1 * e1};
    accS = __builtin_amdgcn_wmma_f32_16x16x4_f32(
        false, Ae, false, ones, (short)0, accS, false, false);
    accW = __builtin_amdgcn_wmma_f32_16x16x4_f32(
        false, Aw, false, ones, (short)0, accW, false, false);
  }

  // C/D layout: lane 0 VGPR rr = D[M=rr][N=0]; lane 16 VGPR rr = D[M=rr+8][N=0]
  float hs = 0.0f;
  #pragma unroll
  for (int rr = 0; rr < 8; ++rr) {
    const float sA = __shfl(accS[rr], 0, 32);
    const float wA = __shfl(accW[rr], 0, 32);
    const float sB = __shfl(accS[rr], 16, 32);
    const float wB = __shfl(accW[rr], 16, 32);
    hs += (__logf(sA) - wA / sA) + (__logf(sB) - wB / sB);
  }
  if (lane == 0) hpart[w] = hs;
  __syncthreads();

  if (threadIdx.x == 0) {
    double l1 = 0.0;
    for (int i = 0; i < SPLIT * MBLK; ++i) l1 += (double)entp[i];
    const double H = (double)hpart[0] + (double)hpart[1]
                   + (double)hpart[2] + (double)hpart[3];
    out[0] = (float)(l1 / ((double)TROWS * (double)MBLK) - H / (double)MBLK);
  }
}

extern "C" void kernel_launch(void* const* d_in, const int* in_sizes, int n_in,
                              void* d_out, int out_size, void* d_ws, size_t ws_size,
                              hipStream_t stream) {
  (void)in_sizes; (void)n_in; (void)out_size; (void)ws_size;
  const float* x = (const float*)d_in[0];
  float* colsum  = (float*)d_ws;                                   // 256 KB
  float* entp    = colsum + (size_t)SPLIT * MBLK * KDIM;           // +1 KB
  float* out     = (float*)d_out;

  entropy_stream_kernel<<<MBLK * SPLIT, 256, 0, stream>>>(x, colsum, entp);
  blockmean_wmma_kernel<<<1, 128, 0, stream>>>(colsum, entp, out);
}